// RandomShiftsAug_39676907880573
// MI455X (gfx1250) — compile-verified
//
#include <hip/hip_runtime.h>
#include <stdint.h>

// ---- problem constants (match reference) ----
#define B_   64
#define C_   9
#define H_   256
#define W_   256
#define PAD_ 4
#define HP_  (H_ + 2 * PAD_)   // 264
#define WP_  (W_ + 2 * PAD_)   // 264

#define ROWS_PER_BLK 4         // output rows per block
#define STAGE_ROWS   6         // staged input rows (covers r0(y)..r1(y+3))

typedef __attribute__((ext_vector_type(4))) unsigned int u32x4;
typedef __attribute__((ext_vector_type(4))) int          i32x4;
typedef __attribute__((ext_vector_type(8))) int          i32x8;

// Toolchain probe: therock-10.0 headers ship the TDM header and use the
// 6-arg builtin; ROCm 7.2 uses the 5-arg form.
#if __has_include(<hip/amd_detail/amd_gfx1250_TDM.h>)
#define TDM_SIX_ARGS 1
#else
#define TDM_SIX_ARGS 0
#endif

static __device__ __forceinline__ void tdm_load_to_lds(u32x4 g0, i32x8 g1,
                                                       i32x4 g2, i32x4 g3) {
#if TDM_SIX_ARGS
  i32x8 gz = {};
  __builtin_amdgcn_tensor_load_to_lds(g0, g1, g2, g3, gz, 0);
#else
  __builtin_amdgcn_tensor_load_to_lds(g0, g1, g2, g3, 0);
#endif
}

// Replicate the reference's float32 coordinate math exactly:
//   g    = linspace(-1+eps, 1-eps, dim)[i]
//   gb   = g + shift * (2/dimP)
//   crd  = ((gb + 1) * dimP - 1) * 0.5
// Bilinear taps on the padded domain, zero weight outside [0, dimP-1],
// padded index j -> original index clamp(j - PAD, 0, dim-1) (edge pad).
static __device__ __forceinline__ void taps(int i, int shift, int dim, int dimP,
                                            float& w0, float& w1,
                                            int& c0, int& c1) {
  const float eps   = 2.0f / (float)dimP;
  const float start = -1.0f + eps;
  const float stop  = 1.0f - eps;
  const float step  = (stop - start) / (float)(dim - 1);
  float g     = start + (float)i * step;
  float gb    = g + (float)shift * (2.0f / (float)dimP);
  float coord = ((gb + 1.0f) * (float)dimP - 1.0f) * 0.5f;
  float f0    = floorf(coord);
  float t     = coord - f0;           // w1
  int   j0    = (int)f0;
  int   j1    = j0 + 1;
  w0 = (j0 >= 0 && j0 <= dimP - 1) ? (1.0f - t) : 0.0f;
  w1 = (j1 >= 0 && j1 <= dimP - 1) ? t : 0.0f;
  c0 = min(max(j0 - PAD_, 0), dim - 1);
  c1 = min(max(j1 - PAD_, 0), dim - 1);
}

__global__ __launch_bounds__(256)
void RandomShiftsAug_kernel(const float* __restrict__ x,
                            const int* __restrict__ shift_x,
                            const int* __restrict__ shift_y,
                            float* __restrict__ out) {
  // Staged tile: 9 channels x 6 rows x 256 floats = 54 KB of the 320 KB WGP LDS.
  __shared__ float tile[C_ * STAGE_ROWS * W_];

  const int bid   = blockIdx.x;            // b * (H/4) + ytile
  const int b     = bid >> 6;
  const int ytile = bid & 63;
  const int y0    = ytile * ROWS_PER_BLK;
  const int tx    = threadIdx.x;           // output x, 0..255

  const int sx = shift_x[b];
  const int sy = shift_y[b];

  // y taps for the 4 output rows (uniform per block -> SALU)
  float wy0[ROWS_PER_BLK], wy1[ROWS_PER_BLK];
  int   ry0[ROWS_PER_BLK], ry1[ROWS_PER_BLK];
#pragma unroll
  for (int r = 0; r < ROWS_PER_BLK; ++r)
    taps(y0 + r, sy, H_, HP_, wy0[r], wy1[r], ry0[r], ry1[r]);

  // Staged window [rstart, rstart+5] covers every clamped row index:
  // r0 is monotone in y and r1(y0+3) <= r0(y0) + 5.
  const int rstart = min(ry0[0], H_ - STAGE_ROWS);
  int o0[ROWS_PER_BLK], o1[ROWS_PER_BLK];
#pragma unroll
  for (int r = 0; r < ROWS_PER_BLK; ++r) {
    o0[r] = ry0[r] - rstart;   // 0..5
    o1[r] = ry1[r] - rstart;   // 0..5
  }

  // ---- TDM: one 54 KB DMA per block (wave 0 issues; EXEC is ignored) ----
  if (tx < 32) {
    const unsigned long long ga =
        (unsigned long long)(uintptr_t)(x + ((size_t)b * C_ * H_ + (size_t)rstart) * W_);
    const unsigned int lds = (unsigned int)(uintptr_t)&tile[0];

    // D# group 0: count=1 | lds_addr | global_addr[56:0] | type=2
    u32x4 g0;
    g0[0] = 1u;
    g0[1] = lds;
    g0[2] = (unsigned int)(ga & 0xffffffffu);
    g0[3] = (unsigned int)(ga >> 32) | (2u << 30);

    // D# group 1: data_size=4B; tensor 256x256x9; tile 256x6x9;
    // dim0_stride = W (row step), dim1_stride = H*W (channel step)
    i32x8 g1;
    g1[0] = (int)(2u << 16);                                  // data_size = 2 (4B)
    g1[1] = (int)((unsigned)W_ << 16);                        // tensor_dim0 lo16
    g1[2] = (int)((unsigned)H_ << 16);                        // d0 hi | tensor_dim1 lo
    g1[3] = (int)((unsigned)W_ << 16);                        // d1 hi | tile_dim0
    g1[4] = (int)((unsigned)STAGE_ROWS | ((unsigned)C_ << 16)); // tile_dim1|tile_dim2
    g1[5] = (int)W_;                                          // dim0_stride[31:0]
    g1[6] = (int)((((unsigned)(H_ * W_)) & 0xffffu) << 16);   // d0s hi | d1s lo16
    g1[7] = (int)(((unsigned)(H_ * W_)) >> 16);               // dim1_stride[47:16]

    // D# group 2/3: tensor_dim2 = C; dim3+/iterate unused
    i32x4 g2;
    g2[0] = C_; g2[1] = 0; g2[2] = 0; g2[3] = 0;
    i32x4 g3 = {};

    tdm_load_to_lds(g0, g1, g2, g3);
  }

  // x taps (per lane) — overlaps with the in-flight DMA
  float wx0, wx1;
  int   cx0, cx1;
  taps(tx, sx, W_, WP_, wx0, wx1, cx0, cx1);

  if (tx < 32) __builtin_amdgcn_s_wait_tensorcnt(0);
  __syncthreads();

  // Compute: 4 output rows x 9 channels, all from LDS.
  float* ob = out + ((size_t)b * C_ * H_ + (size_t)y0) * W_ + tx;

#pragma unroll
  for (int c = 0; c < C_; ++c) {
    const float* tc = &tile[c * (STAGE_ROWS * W_)];
    float* oc = ob + (size_t)c * H_ * W_;
#pragma unroll
    for (int r = 0; r < ROWS_PER_BLK; ++r) {
      const float* q0 = tc + o0[r] * W_;
      const float* q1 = tc + o1[r] * W_;
      // match reference order: horizontal lerp per padded row, then vertical
      float h0 = q0[cx0] * wx0 + q0[cx1] * wx1;
      float h1 = q1[cx0] * wx0 + q1[cx1] * wx1;
      float v  = h0 * wy0[r] + h1 * wy1[r];
      __builtin_nontemporal_store(v, oc + (size_t)r * W_);
    }
  }
}

extern "C" void kernel_launch(void* const* d_in, const int* in_sizes, int n_in,
                              void* d_out, int out_size, void* d_ws, size_t ws_size,
                              hipStream_t stream) {
  (void)in_sizes; (void)n_in; (void)out_size; (void)d_ws; (void)ws_size;
  const float* x  = (const float*)d_in[0];
  const int*   sx = (const int*)d_in[1];
  const int*   sy = (const int*)d_in[2];
  float*       o  = (float*)d_out;

  dim3 grid(B_ * (H_ / ROWS_PER_BLK));   // 4096 blocks: one per (b, 4-row tile)
  dim3 block(256);                       // 8 wave32
  hipLaunchKernelGGL(RandomShiftsAug_kernel, grid, block, 0, stream, x, sx, sy, o);
}